// GATom_76544907149765
// MI455X (gfx1250) — compile-verified
//
#include <hip/hip_runtime.h>
#include <hip/hip_bf16.h>
#include <math.h>

// ---------------------------------------------------------------------------
// GATv2 GNN pipeline for MI455X (gfx1250, wave32).
// Memory-bound (~2 GB traffic vs ~25 GFLOP @ 23.3 TB/s) => fp32 end-to-end.
// Dense GEMMs: V_WMMA_F32_16X16X4_F32, 64x64x32 block tiles, float4 (b128)
// staged through LDS, global_prefetch_b8 on the next K tile.
// Graph phases: wave-per-edge b128 gathers + f32 global atomics for
// segment softmax / segment sum.
// ---------------------------------------------------------------------------

typedef float v2f __attribute__((ext_vector_type(2)));
typedef float v8f __attribute__((ext_vector_type(8)));

#define TB 256      // threads per block (8 wave32)
#define BM 64       // block tile M (2 wave rows x 2 sub-tiles)
#define BN 64       // block tile N (4 wave cols)
#define BK 32       // K step (8 x k4 per stage)

__device__ __forceinline__ float siluf(float v) {
    return v / (1.0f + __expf(-v));
}

// ---------------- WMMA fp32 GEMM: C = act(A[M,K] @ B[K,N] + bias) ----------
template<int ACT>   // 0 = none, 1 = silu
__global__ __launch_bounds__(TB)
void gemm_wmma(const float* __restrict__ A, const float* __restrict__ B,
               const float* __restrict__ bias, float* __restrict__ C,
               int M, int K, int N)
{
    __shared__ float As[BM][BK + 1];   // +1 pad: row reads conflict-free
    __shared__ float Bs[BK][BN];

    const int tid  = threadIdx.x;
    const int lane = tid & 31;
    const int wave = tid >> 5;
    const int waveRow = wave >> 2;     // 0..1 -> 32 rows each
    const int waveCol = wave & 3;      // 0..3
    const int blockM = blockIdx.y * BM;
    const int blockN = blockIdx.x * BN;
    const bool vecA  = (K & 3) == 0;

    v8f acc0 = {}, acc1 = {};

    const int m0    = waveRow * 32 + (lane & 15);        // sub-tile 0 row
    const int nloc  = waveCol * 16 + (lane & 15);
    const int khalf = (lane >> 4) * 2;                   // lanes 0-15: K{0,1}; 16-31: K{2,3}

    for (int k0 = 0; k0 < K; k0 += BK) {
        // ---- stage A tile: 64x32 = 512 float4 slots, 2 per thread ----
        #pragma unroll
        for (int s = 0; s < 2; ++s) {
            int slot = tid + s * TB;
            int m  = slot >> 3;                 // /(BK/4)
            int k4 = (slot & 7) << 2;
            int gm = blockM + m, gk = k0 + k4;
            float4 v = make_float4(0.f, 0.f, 0.f, 0.f);
            if (gm < M) {
                const float* ap = A + (long)gm * K + gk;
                if (vecA && gk + 3 < K) {
                    v = *(const float4*)ap;
                    if (gk + BK + 3 < K) __builtin_prefetch(ap + BK, 0, 0);
                } else {
                    if (gk     < K) v.x = ap[0];
                    if (gk + 1 < K) v.y = ap[1];
                    if (gk + 2 < K) v.z = ap[2];
                    if (gk + 3 < K) v.w = ap[3];
                }
            }
            As[m][k4] = v.x; As[m][k4+1] = v.y; As[m][k4+2] = v.z; As[m][k4+3] = v.w;
        }
        // ---- stage B tile: 32x64 = 512 float4 slots, 2 per thread ----
        #pragma unroll
        for (int s = 0; s < 2; ++s) {
            int slot = tid + s * TB;
            int k  = slot >> 4;                 // /(BN/4)
            int n4 = (slot & 15) << 2;
            int gk = k0 + k, gn = blockN + n4;
            float4 v = make_float4(0.f, 0.f, 0.f, 0.f);
            if (gk < K) {
                const float* bp = B + (long)gk * N + gn;
                if (gn + 3 < N) {
                    v = *(const float4*)bp;
                    if (gk + BK < K) __builtin_prefetch(bp + (long)BK * N, 0, 0);
                } else {
                    if (gn     < N) v.x = bp[0];
                    if (gn + 1 < N) v.y = bp[1];
                    if (gn + 2 < N) v.z = bp[2];
                }
            }
            Bs[k][n4] = v.x; Bs[k][n4+1] = v.y; Bs[k][n4+2] = v.z; Bs[k][n4+3] = v.w;
        }
        __syncthreads();

        #pragma unroll
        for (int kk = 0; kk < BK; kk += 4) {
            v2f b;
            b.x = Bs[kk + khalf][nloc];
            b.y = Bs[kk + khalf + 1][nloc];
            v2f a;
            a.x = As[m0][kk + khalf];
            a.y = As[m0][kk + khalf + 1];
            acc0 = __builtin_amdgcn_wmma_f32_16x16x4_f32(
                       false, a, false, b, (short)0, acc0, false, false);
            a.x = As[m0 + 16][kk + khalf];
            a.y = As[m0 + 16][kk + khalf + 1];
            acc1 = __builtin_amdgcn_wmma_f32_16x16x4_f32(
                       false, a, false, b, (short)0, acc1, false, false);
        }
        __syncthreads();
    }

    // C/D layout: VGPR r -> row r (lanes 0-15) / row r+8 (lanes 16-31)
    const int nglob = blockN + nloc;
    const int mbase = blockM + waveRow * 32 + ((lane >> 4) ? 8 : 0);
    if (nglob < N) {
        const float bv = bias ? bias[nglob] : 0.0f;
        #pragma unroll
        for (int r = 0; r < 8; ++r) {
            int mg = mbase + r;
            if (mg < M) {
                float v = acc0[r] + bv;
                if (ACT == 1) v = siluf(v);
                C[(long)mg * N + nglob] = v;
            }
            int mg1 = mg + 16;
            if (mg1 < M) {
                float v = acc1[r] + bv;
                if (ACT == 1) v = siluf(v);
                C[(long)mg1 * N + nglob] = v;
            }
        }
    }
}

// ---------------- graph attention kernels ----------------------------------
__device__ __forceinline__ void atomicMaxFloat(float* addr, float val) {
    if (val >= 0.0f) atomicMax((int*)addr, __float_as_int(val));
    else             atomicMin((unsigned int*)addr, __float_as_uint(val));
}

// wave-per-edge: logit[e,h] = sum_c att[h,c] * leaky_relu(xl[src]+xr[dst](+ep))
// one float4 (b128) per lane covers the full HC<=128 row.
__global__ __launch_bounds__(TB)
void edge_logits(const float* __restrict__ xl, const float* __restrict__ xr,
                 const float* __restrict__ ep, const int* __restrict__ src,
                 const int* __restrict__ dst, const float* __restrict__ att,
                 float* __restrict__ logit, int E_, int HC)
{
    int gw   = (blockIdx.x * TB + threadIdx.x) >> 5;
    int lane = threadIdx.x & 31;
    if (gw >= E_) return;
    int s = src ? src[gw] : gw;
    int d = dst[gw];
    int idx = lane << 2;
    float a0 = 0.0f, a1 = 0.0f;
    if (idx < HC) {
        float4 xlv = *(const float4*)(xl + (long)s * HC + idx);
        float4 xrv = *(const float4*)(xr + (long)d * HC + idx);
        float4 epv = make_float4(0.f, 0.f, 0.f, 0.f);
        if (ep) epv = *(const float4*)(ep + (long)gw * HC + idx);
        float4 atv = *(const float4*)(att + idx);
        float m, z, c = 0.0f;
        m = xlv.x + xrv.x + epv.x; z = m > 0.f ? m : 0.01f * m; c += z * atv.x;
        m = xlv.y + xrv.y + epv.y; z = m > 0.f ? m : 0.01f * m; c += z * atv.y;
        m = xlv.z + xrv.z + epv.z; z = m > 0.f ? m : 0.01f * m; c += z * atv.z;
        m = xlv.w + xrv.w + epv.w; z = m > 0.f ? m : 0.01f * m; c += z * atv.w;
        if (idx < 64) a0 = c; else a1 = c;   // 4 consecutive cols share a head
    }
    #pragma unroll
    for (int off = 16; off; off >>= 1) {
        a0 += __shfl_xor(a0, off, 32);
        a1 += __shfl_xor(a1, off, 32);
    }
    if (lane == 0) {
        int H_ = HC >> 6;
        logit[(long)gw * H_] = a0;
        if (H_ > 1) logit[(long)gw * H_ + 1] = a1;
    }
}

__global__ void seg_max(const float* __restrict__ logit, const int* __restrict__ dst,
                        float* __restrict__ maxb, long n, int H_)
{
    long i = (long)blockIdx.x * blockDim.x + threadIdx.x;
    if (i >= n) return;
    long eid = i / H_; int h = (int)(i % H_);
    atomicMaxFloat(&maxb[(long)dst[eid] * H_ + h], logit[i]);
}

__global__ void seg_expsum(float* __restrict__ logit, const int* __restrict__ dst,
                           const float* __restrict__ maxb, float* __restrict__ sumb,
                           long n, int H_)
{
    long i = (long)blockIdx.x * blockDim.x + threadIdx.x;
    if (i >= n) return;
    long eid = i / H_; int h = (int)(i % H_);
    int d = dst[eid];
    float ex = __expf(logit[i] - maxb[(long)d * H_ + h]);
    logit[i] = ex;                                        // in-place -> exp value
    atomicAdd(&sumb[(long)d * H_ + h], ex);
}

// wave-per-edge: out[dst] += xl[src] * alpha (b128 gather, scalar atomics)
__global__ __launch_bounds__(TB)
void seg_scatter(const float* __restrict__ xl, const float* __restrict__ ex,
                 const float* __restrict__ sumb, const int* __restrict__ src,
                 const int* __restrict__ dst, float* __restrict__ out,
                 int E_, int HC)
{
    int gw   = (blockIdx.x * TB + threadIdx.x) >> 5;
    int lane = threadIdx.x & 31;
    if (gw >= E_) return;
    int s = src ? src[gw] : gw;
    int d = dst[gw];
    int H_ = HC >> 6;
    int idx = lane << 2;
    if (idx >= HC) return;
    int h = idx >> 6;
    float alpha = ex[(long)gw * H_ + h] / (sumb[(long)d * H_ + h] + 1e-16f);
    float4 xlv = *(const float4*)(xl + (long)s * HC + idx);
    float* o = out + (long)d * HC + idx;
    atomicAdd(o + 0, xlv.x * alpha);
    atomicAdd(o + 1, xlv.y * alpha);
    atomicAdd(o + 2, xlv.z * alpha);
    atomicAdd(o + 3, xlv.w * alpha);
}

// ---------------- elementwise / misc ---------------------------------------
__global__ void fill_k(float* __restrict__ p, float v, long n) {
    long i = (long)blockIdx.x * blockDim.x + threadIdx.x;
    if (i < n) p[i] = v;
}

// out[i] = [ a[i] (wa cols) | b[i] + bbias (wb cols) ], float4 granularity
__global__ void concat_bias(const float* __restrict__ a, int wa,
                            const float* __restrict__ b, const float* __restrict__ bbias,
                            int wb, float* __restrict__ out, long rows)
{
    int w4 = (wa + wb) >> 2;
    long i = (long)blockIdx.x * blockDim.x + threadIdx.x;
    if (i >= rows * w4) return;
    long r = i / w4; int c = (int)(i % w4) << 2;
    float4 v;
    if (c < wa) {
        v = *(const float4*)(a + r * wa + c);
    } else {
        int cb = c - wa;
        v = *(const float4*)(b + r * wb + cb);
        if (bbias) {
            const float4 bb = *(const float4*)(bbias + cb);
            v.x += bb.x; v.y += bb.y; v.z += bb.z; v.w += bb.w;
        }
    }
    *(float4*)(out + i * 4) = v;
}

__device__ __forceinline__ float eluf(float g) {
    return g > 0.0f ? g : (__expf(g) - 1.0f);
}

// GLU: y [rows, 2C] -> out [rows, C] = v * elu(g), float4 granularity
__global__ void glu_k(const float* __restrict__ y, float* __restrict__ out,
                      long rows, int C_)
{
    int c4 = C_ >> 2;
    long i = (long)blockIdx.x * blockDim.x + threadIdx.x;
    if (i >= rows * c4) return;
    long r = i / c4; int c = (int)(i % c4) << 2;
    float4 v = *(const float4*)(y + r * 2 * C_ + c);
    float4 g = *(const float4*)(y + r * 2 * C_ + C_ + c);
    float4 o;
    o.x = v.x * eluf(g.x); o.y = v.y * eluf(g.y);
    o.z = v.z * eluf(g.z); o.w = v.w * eluf(g.w);
    *(float4*)(out + i * 4) = o;
}

// ---------------- DiffGroupNorm --------------------------------------------
#define NGRP 10
__global__ void dgn_assign(const float* __restrict__ x, const float* __restrict__ W,
                           float* __restrict__ s, int n)
{
    int i = blockIdx.x * blockDim.x + threadIdx.x;
    if (i >= n) return;
    float v[NGRP];
    float mx = -1e30f;
    #pragma unroll
    for (int g = 0; g < NGRP; ++g) v[g] = 0.0f;
    for (int c = 0; c < 64; c += 4) {
        float4 xv = *(const float4*)(x + (long)i * 64 + c);
        #pragma unroll
        for (int g = 0; g < NGRP; ++g) {
            v[g] += xv.x * W[(c    ) * NGRP + g] + xv.y * W[(c + 1) * NGRP + g]
                  + xv.z * W[(c + 2) * NGRP + g] + xv.w * W[(c + 3) * NGRP + g];
        }
    }
    #pragma unroll
    for (int g = 0; g < NGRP; ++g) mx = fmaxf(mx, v[g]);
    float sum = 0.0f;
    #pragma unroll
    for (int g = 0; g < NGRP; ++g) { v[g] = __expf(v[g] - mx); sum += v[g]; }
    #pragma unroll
    for (int g = 0; g < NGRP; ++g) s[(long)i * NGRP + g] = v[g] / sum;
}

// one block per (group,channel) column j of o = s (x) x ; batch stats over n rows
__global__ __launch_bounds__(TB)
void dgn_stats(const float* __restrict__ x, const float* __restrict__ s,
               float* __restrict__ mu, float* __restrict__ rstd, int n)
{
    int j = blockIdx.x;                 // 0..639
    int g = j / 64, c = j % 64;
    float sum = 0.0f, sq = 0.0f;
    for (int i = threadIdx.x; i < n; i += TB) {
        float o = s[(long)i * NGRP + g] * x[(long)i * 64 + c];
        sum += o; sq += o * o;
    }
    __shared__ float ssum[TB], ssq[TB];
    ssum[threadIdx.x] = sum; ssq[threadIdx.x] = sq;
    __syncthreads();
    for (int off = TB / 2; off; off >>= 1) {
        if (threadIdx.x < off) {
            ssum[threadIdx.x] += ssum[threadIdx.x + off];
            ssq[threadIdx.x]  += ssq[threadIdx.x + off];
        }
        __syncthreads();
    }
    if (threadIdx.x == 0) {
        float m = ssum[0] / (float)n;
        float var = ssq[0] / (float)n - m * m;
        mu[j]   = m;
        rstd[j] = rsqrtf(var + 1e-5f);
    }
}

__global__ void dgn_apply(float* __restrict__ x, const float* __restrict__ s,
                          const float* __restrict__ mu, const float* __restrict__ rstd,
                          long n)
{
    long i = (long)blockIdx.x * blockDim.x + threadIdx.x;
    if (i >= n * 64) return;
    long r = i / 64; int c = (int)(i % 64);
    float xv = x[i];
    float acc = 0.0f;
    #pragma unroll
    for (int g = 0; g < NGRP; ++g) {
        int j = g * 64 + c;
        acc += (s[r * NGRP + g] * xv - mu[j]) * rstd[j];
    }
    x[i] = xv + 0.01f * acc;
}

// ---------------- pooling & head -------------------------------------------
__global__ void pool_sum(const float* __restrict__ h, const int* __restrict__ batch,
                         float* __restrict__ pooled, long n)
{
    long i = (long)blockIdx.x * blockDim.x + threadIdx.x;
    if (i >= n * 64) return;
    long r = i / 64; int c = (int)(i % 64);
    atomicAdd(&pooled[(long)batch[r] * 64 + c], h[i]);
}

__global__ void relu_k(float* __restrict__ x, long n) {
    long i = (long)blockIdx.x * blockDim.x + threadIdx.x;
    if (i < n) x[i] = fmaxf(x[i], 0.0f);
}

__global__ void out_head(const float* __restrict__ z, const float* __restrict__ Wout,
                         const float* __restrict__ bout, float* __restrict__ out, int G_)
{
    int g = blockIdx.x * blockDim.x + threadIdx.x;
    if (g >= G_) return;
    float acc = bout[0];
    for (int c = 0; c < 64; ++c) acc += z[(long)g * 64 + c] * Wout[c];
    out[g] = acc;
}

// ---------------------------------------------------------------------------
static inline int cdiv(long a, long b) { return (int)((a + b - 1) / b); }

extern "C" void kernel_launch(void* const* d_in, const int* in_sizes, int n_in,
                              void* d_out, int out_size, void* d_ws, size_t ws_size,
                              hipStream_t stream)
{
    (void)n_in; (void)ws_size;
    const int N  = in_sizes[0] / 92;      // 50000
    const int E  = in_sizes[1] / 50;      // 500000
    const int G  = out_size;              // 512
    const int H  = 2, HID = 64, HC = H * HID;

    const float* x_in   = (const float*)d_in[0];
    const float* eattr  = (const float*)d_in[1];
    const int*   ei     = (const int*)d_in[2];
    const int*   src    = ei;
    const int*   dst    = ei + E;
    const int*   batch  = (const int*)d_in[3];
    const float* Wn  = (const float*)d_in[4];  const float* bn  = (const float*)d_in[5];
    const float* Wep = (const float*)d_in[6];  const float* bep = (const float*)d_in[7];
    const float* cWl = (const float*)d_in[8];  const float* cbl = (const float*)d_in[9];
    const float* cWr = (const float*)d_in[10]; const float* cbr = (const float*)d_in[11];
    const float* cWe = (const float*)d_in[12];
    const float* catt= (const float*)d_in[13]; const float* cbias=(const float*)d_in[14];
    const float* gluW= (const float*)d_in[15]; const float* glub =(const float*)d_in[16];
    const float* normW=(const float*)d_in[17];
    const float* gWl = (const float*)d_in[18]; const float* gbl = (const float*)d_in[19];
    const float* gWr = (const float*)d_in[20]; const float* gbr = (const float*)d_in[21];
    const float* gatt= (const float*)d_in[22]; const float* gbias=(const float*)d_in[23];
    const float* ggluW=(const float*)d_in[24]; const float* gglub=(const float*)d_in[25];
    const float* gnormW=(const float*)d_in[26];
    const float* W1  = (const float*)d_in[27]; const float* b1 = (const float*)d_in[28];
    const float* W2  = (const float*)d_in[29]; const float* b2 = (const float*)d_in[30];
    const float* Wout= (const float*)d_in[31]; const float* bout=(const float*)d_in[32];

    // ---- workspace carve (floats) ----
    float* ws = (float*)d_ws;
    long off = 0;
    auto alloc = [&](long nf) { float* p = ws + off; off += nf; return p; };
    float* e_buf  = alloc((long)E * HID);   // silu(edge_attr @ Wep)
    float* ep_buf = alloc((long)E * HC);    // e @ cWe[l]  (reused per layer)
    float* h_buf  = alloc((long)N * HID);
    float* xl_buf = alloc((long)N * HC);
    float* xr_buf = alloc((long)N * HC);
    float* a_buf  = alloc((long)N * HC);    // attention output (embedding layers)
    float* cat_buf= alloc((long)N * (HID + HC));
    float* y_buf  = alloc((long)N * HC);    // GLU pre-activation
    float* log_buf= alloc((long)E * H);     // logits -> exp (in place)
    float* max_buf= alloc((long)N * H);
    float* sum_buf= alloc((long)N * H);
    float* s_buf  = alloc((long)N * NGRP);
    float* mu_buf = alloc(NGRP * 64);
    float* rs_buf = alloc(NGRP * 64);
    float* pooled = alloc((long)G * HID);
    float* gxr    = alloc((long)G * HID);
    float* gout_b = alloc((long)G * HID);
    float* gcat   = alloc((long)G * 2 * HID);
    float* gy     = alloc((long)G * 2 * HID);
    float* gz     = alloc((long)G * HID);
    float* z1     = alloc((long)G * HID);
    float* z2     = alloc((long)G * HID);

    dim3 tb(TB);

    // ---- pre-conv linears ----
    gemm_wmma<1><<<dim3(1, cdiv(N, BM)), tb, 0, stream>>>(x_in,  Wn,  bn,  h_buf, N, 92, HID);
    gemm_wmma<1><<<dim3(1, cdiv(E, BM)), tb, 0, stream>>>(eattr, Wep, bep, e_buf, E, 50, HID);

    // ---- two embedding GATv2 layers ----
    for (int l = 0; l < 2; ++l) {
        const float* Wl = cWl + (long)l * HID * HC;
        const float* Wr = cWr + (long)l * HID * HC;
        const float* We = cWe + (long)l * HID * HC;

        gemm_wmma<0><<<dim3(2, cdiv(E, BM)), tb, 0, stream>>>(e_buf, We, nullptr, ep_buf, E, HID, HC);
        gemm_wmma<0><<<dim3(2, cdiv(N, BM)), tb, 0, stream>>>(h_buf, Wl, cbl + l*HC, xl_buf, N, HID, HC);
        gemm_wmma<0><<<dim3(2, cdiv(N, BM)), tb, 0, stream>>>(h_buf, Wr, cbr + l*HC, xr_buf, N, HID, HC);

        fill_k<<<cdiv((long)N*H, TB),  tb, 0, stream>>>(max_buf, -INFINITY, (long)N*H);
        fill_k<<<cdiv((long)N*H, TB),  tb, 0, stream>>>(sum_buf, 0.0f,      (long)N*H);
        fill_k<<<cdiv((long)N*HC, TB), tb, 0, stream>>>(a_buf,   0.0f,      (long)N*HC);

        edge_logits<<<cdiv((long)E*32, TB), tb, 0, stream>>>(xl_buf, xr_buf, ep_buf, src, dst,
                                                             catt + l*H*HID, log_buf, E, HC);
        seg_max   <<<cdiv((long)E*H, TB), tb, 0, stream>>>(log_buf, dst, max_buf, (long)E*H, H);
        seg_expsum<<<cdiv((long)E*H, TB), tb, 0, stream>>>(log_buf, dst, max_buf, sum_buf, (long)E*H, H);
        seg_scatter<<<cdiv((long)E*32, TB), tb, 0, stream>>>(xl_buf, log_buf, sum_buf, src, dst,
                                                             a_buf, E, HC);

        concat_bias<<<cdiv((long)N*(HID+HC)/4, TB), tb, 0, stream>>>(h_buf, HID, a_buf,
                                                                     cbias + l*HC, HC, cat_buf, N);
        gemm_wmma<0><<<dim3(2, cdiv(N, BM)), tb, 0, stream>>>(cat_buf, gluW + (long)l*(HID+HC)*HC,
                                                              glub + l*HC, y_buf, N, HID+HC, HC);
        glu_k<<<cdiv((long)N*HID/4, TB), tb, 0, stream>>>(y_buf, h_buf, N, HID);

        dgn_assign<<<cdiv(N, TB), tb, 0, stream>>>(h_buf, normW + l*HID*NGRP, s_buf, N);
        dgn_stats <<<NGRP*64, tb, 0, stream>>>(h_buf, s_buf, mu_buf, rs_buf, N);
        dgn_apply <<<cdiv((long)N*HID, TB), tb, 0, stream>>>(h_buf, s_buf, mu_buf, rs_buf, N);
    }

    // ---- pooling ----
    fill_k<<<cdiv((long)G*HID, TB), tb, 0, stream>>>(pooled, 0.0f, (long)G*HID);
    pool_sum<<<cdiv((long)N*HID, TB), tb, 0, stream>>>(h_buf, batch, pooled, N);
    relu_k<<<cdiv((long)G*HID, TB), tb, 0, stream>>>(pooled, (long)G*HID);

    // ---- global bipartite GATv2 (heads=1) : node -> graph ----
    gemm_wmma<0><<<dim3(1, cdiv(N, BM)), tb, 0, stream>>>(h_buf,  gWl, gbl, xl_buf, N, HID, HID);
    gemm_wmma<0><<<dim3(1, cdiv(G, BM)), tb, 0, stream>>>(pooled, gWr, gbr, gxr,    G, HID, HID);

    fill_k<<<cdiv((long)G, TB), tb, 0, stream>>>(max_buf, -INFINITY, (long)G);
    fill_k<<<cdiv((long)G, TB), tb, 0, stream>>>(sum_buf, 0.0f,      (long)G);
    fill_k<<<cdiv((long)G*HID, TB), tb, 0, stream>>>(gout_b, 0.0f,   (long)G*HID);

    edge_logits<<<cdiv((long)N*32, TB), tb, 0, stream>>>(xl_buf, gxr, nullptr, nullptr, batch,
                                                         gatt, log_buf, N, HID);
    seg_max   <<<cdiv((long)N, TB), tb, 0, stream>>>(log_buf, batch, max_buf, (long)N, 1);
    seg_expsum<<<cdiv((long)N, TB), tb, 0, stream>>>(log_buf, batch, max_buf, sum_buf, (long)N, 1);
    seg_scatter<<<cdiv((long)N*32, TB), tb, 0, stream>>>(xl_buf, log_buf, sum_buf, nullptr, batch,
                                                         gout_b, N, HID);

    concat_bias<<<cdiv((long)G*2*HID/4, TB), tb, 0, stream>>>(pooled, HID, gout_b, gbias, HID, gcat, G);
    gemm_wmma<0><<<dim3(2, cdiv(G, BM)), tb, 0, stream>>>(gcat, ggluW, gglub, gy, G, 2*HID, 2*HID);
    glu_k<<<cdiv((long)G*HID/4, TB), tb, 0, stream>>>(gy, gz, G, HID);

    dgn_assign<<<cdiv(G, TB), tb, 0, stream>>>(gz, gnormW, s_buf, G);
    dgn_stats <<<NGRP*64, tb, 0, stream>>>(gz, s_buf, mu_buf, rs_buf, G);
    dgn_apply <<<cdiv((long)G*HID, TB), tb, 0, stream>>>(gz, s_buf, mu_buf, rs_buf, G);

    // ---- MLP head ----
    gemm_wmma<1><<<dim3(1, cdiv(G, BM)), tb, 0, stream>>>(gz, W1, b1, z1, G, HID, HID);
    gemm_wmma<1><<<dim3(1, cdiv(G, BM)), tb, 0, stream>>>(z1, W2, b2, z2, G, HID, HID);
    out_head<<<cdiv(G, TB), tb, 0, stream>>>(z2, Wout, bout, (float*)d_out, G);
}